// SKANLinear_ranged_59313498358104
// MI455X (gfx1250) — compile-verified
//
#include <hip/hip_runtime.h>

#define BLOCK_THREADS 256

namespace {
constexpr int kB   = 1024;
constexpr int kIn  = 512;
constexpr int kInP = 513;   // IN + bias column
constexpr int kOut = 512;

constexpr int BT  = 64;     // batch tile rows
constexpr int OT  = 32;     // output tile rows
constexpr int CH  = 16;     // i-chunk length (floats)
constexpr int XS  = 20;     // LDS row stride: 16 data + 4 pad floats (TDM pad 16dw/4dw)
constexpr int NCH = kIn / CH;
}

// ---------------------------------------------------------------------------
// CDNA5 Tensor Data Mover path (gfx1250): DMA 2D tiles global->LDS, tracked
// with TENSORcnt. Descriptor packed per CDNA5 ISA ch.8 (D# groups 0..3).
// This toolchain exposes the 6-arg builtin form:
//   (u32x4 g0, i32x8 g1, i32x4 g2, i32x4 g3, i32x8 g4, i32 cpol)
// ---------------------------------------------------------------------------
#if defined(__HIP_DEVICE_COMPILE__) && defined(__has_builtin)
#if __has_builtin(__builtin_amdgcn_tensor_load_to_lds) && __has_builtin(__builtin_amdgcn_s_wait_tensorcnt)
#define SKAN_USE_TDM 1
#endif
#endif
#ifndef SKAN_USE_TDM
#define SKAN_USE_TDM 0
#endif

#if SKAN_USE_TDM
typedef unsigned int u32x4 __attribute__((ext_vector_type(4)));
typedef int          i32x8 __attribute__((ext_vector_type(8)));
typedef int          i32x4 __attribute__((ext_vector_type(4)));

// Load a rows x CH float tile (row stride row_stride_elts in global memory)
// into LDS at lds_byte_off, padding each 16-dword row with 4 dwords so the
// LDS row stride is XS=20 floats (bank-conflict-free for our read pattern).
__device__ __forceinline__ void tdm_load_tile_2d(const float* gsrc,
                                                 unsigned lds_byte_off,
                                                 unsigned rows,
                                                 unsigned row_stride_elts)
{
  const unsigned long long ga = (unsigned long long)(const void*)gsrc;

  u32x4 g0;
  g0.x = 1u;                                     // count=1 (valid), user mode
  g0.y = lds_byte_off;                           // lds_addr (bytes)
  g0.z = (unsigned)(ga & 0xffffffffull);         // global_addr[31:0]
  g0.w = (unsigned)((ga >> 32) & 0x01ffffffull)  // global_addr[56:32]
       | (2u << 30);                             // type = 2 ("image")

  const unsigned td0 = (unsigned)CH;             // tensor_dim0 == tile_dim0
  const unsigned td1 = rows;                     // tensor_dim1 == tile_dim1
  i32x8 g1;
  g1[0] = (int)((2u << 16)                       // data_size = 2 -> 4 bytes
              | (1u << 20)                       // pad_enable
              | (3u << 22)                       // pad_interval code 3 -> 16 dw
              | (3u << 25));                     // pad_amount  code 3 -> 4 dw
  g1[1] = (int)((td0 & 0xffffu) << 16);          // tensor_dim0[15:0] @ bit48
  g1[2] = (int)((td0 >> 16) | ((td1 & 0xffffu) << 16)); // dim0 hi | dim1 lo
  g1[3] = (int)((td1 >> 16) | ((unsigned)CH << 16));    // dim1 hi | tile_dim0
  g1[4] = (int)(rows & 0xffffu);                 // tile_dim1 (tile_dim2 = 0)
  g1[5] = (int)row_stride_elts;                  // tensor_dim0_stride[31:0]
  g1[6] = 0;                                     // stride hi, dim1_stride lo
  g1[7] = 0;

  const i32x4 gz4 = {0, 0, 0, 0};                // groups 2/3 unused (2D tile)
  const i32x8 gz8 = {0, 0, 0, 0, 0, 0, 0, 0};    // trailing group (unused)
  __builtin_amdgcn_tensor_load_to_lds(g0, g1, gz4, gz4, gz8, 0);
}
#endif

// ---------------------------------------------------------------------------
// y[b,o] = sum_i W[o,i] * sin(F[o,i] * xb[b,i]),  xb[:,512] = 1 (bias col).
// Tile: 64 b x 32 o per block; thread micro-tile 4 b x 2 o (8 accumulators).
// i-loop is chunked by 16 and double-buffered through LDS via TDM DMA.
// ---------------------------------------------------------------------------
__global__ __launch_bounds__(BLOCK_THREADS)
void skan_sin_kernel(const float* __restrict__ x,
                     const float* __restrict__ wgt,
                     const float* __restrict__ frq,
                     float* __restrict__ out)
{
  __shared__ float xsh[2][BT * XS];
  __shared__ float fsh[2][OT * XS];
  __shared__ float wsh[2][OT * XS];

  const int tid   = threadIdx.x;
  const int tx    = tid & 15;    // o lane: owns o-local {tx, tx+16}
  const int tyb   = tid >> 4;    // b lane: owns b-local {tyb, +16, +32, +48}
  const int oBase = blockIdx.x * OT;
  const int bBase = blockIdx.y * BT;

  float acc[4][2] = {};

  auto stage = [&](int ch, int buf) {
    const int i0 = ch * CH;
#if SKAN_USE_TDM
    if (tid == 0) {  // TDM ignores EXEC; one wave issues 3 DMA tile loads
      tdm_load_tile_2d(x   + (size_t)bBase * kIn  + i0,
                       (unsigned)(size_t)(void*)&xsh[buf][0], BT, kIn);
      tdm_load_tile_2d(frq + (size_t)oBase * kInP + i0,
                       (unsigned)(size_t)(void*)&fsh[buf][0], OT, kInP);
      tdm_load_tile_2d(wgt + (size_t)oBase * kInP + i0,
                       (unsigned)(size_t)(void*)&wsh[buf][0], OT, kInP);
    }
#else
    {
      // x tile: 64 rows x 16, 16B-aligned -> one float4 per thread
      const int r  = tid >> 2;
      const int c4 = (tid & 3) << 2;
      const float4 v = *(const float4*)(x + (size_t)(bBase + r) * kIn + i0 + c4);
      *(float4*)&xsh[buf][r * XS + c4] = v;
      // frq/wgt tiles: rows have odd stride 513 -> scalar copies
      const int r2 = tid >> 3;
      const int c  = tid & 7;
      const size_t ro = (size_t)(oBase + r2) * kInP + (size_t)i0;
      fsh[buf][r2 * XS + c]     = frq[ro + c];
      fsh[buf][r2 * XS + c + 8] = frq[ro + c + 8];
      wsh[buf][r2 * XS + c]     = wgt[ro + c];
      wsh[buf][r2 * XS + c + 8] = wgt[ro + c + 8];
    }
#endif
  };

  stage(0, 0);

  for (int ch = 0; ch < NCH; ++ch) {
    const int buf = ch & 1;
#if SKAN_USE_TDM
    __builtin_amdgcn_s_wait_tensorcnt(0);   // this chunk's DMA complete
#endif
    __syncthreads();                        // publish buf; prev compute done
    if (ch + 1 < NCH) stage(ch + 1, buf ^ 1);  // DMA next chunk during compute

    #pragma unroll
    for (int c = 0; c < CH; ++c) {
      const float f0 = fsh[buf][tx * XS + c];
      const float f1 = fsh[buf][(tx + 16) * XS + c];
      const float w0 = wsh[buf][tx * XS + c];
      const float w1 = wsh[buf][(tx + 16) * XS + c];
      #pragma unroll
      for (int k = 0; k < 4; ++k) {
        const float xv = xsh[buf][(tyb + (k << 4)) * XS + c];
        acc[k][0] = fmaf(w0, __sinf(f0 * xv), acc[k][0]);  // v_sin_f32 path
        acc[k][1] = fmaf(w1, __sinf(f1 * xv), acc[k][1]);
      }
    }
  }

  // Bias column i = 512 (xb == 1): same for every b of this thread.
  const int o0 = oBase + tx, o1 = o0 + 16;
  const float bias0 = wgt[(size_t)o0 * kInP + kIn] * __sinf(frq[(size_t)o0 * kInP + kIn]);
  const float bias1 = wgt[(size_t)o1 * kInP + kIn] * __sinf(frq[(size_t)o1 * kInP + kIn]);

  #pragma unroll
  for (int k = 0; k < 4; ++k) {
    const int b = bBase + tyb + (k << 4);
    out[(size_t)b * kOut + o0] = acc[k][0] + bias0;
    out[(size_t)b * kOut + o1] = acc[k][1] + bias1;
  }
}

extern "C" void kernel_launch(void* const* d_in, const int* in_sizes, int n_in,
                              void* d_out, int out_size, void* d_ws, size_t ws_size,
                              hipStream_t stream) {
  (void)in_sizes; (void)n_in; (void)d_ws; (void)ws_size; (void)out_size;
  const float* x   = (const float*)d_in[0];
  const float* wgt = (const float*)d_in[1];
  const float* frq = (const float*)d_in[2];
  float* out = (float*)d_out;

  dim3 grid(kOut / OT, kB / BT);   // 16 x 16 = 256 blocks, 256 threads each
  hipLaunchKernelGGL(skan_sin_kernel, grid, dim3(BLOCK_THREADS), 0, stream,
                     x, wgt, frq, out);
}